// VQVAE_2379411882289
// MI455X (gfx1250) — compile-verified
//
#include <hip/hip_runtime.h>
#include <hip/hip_bf16.h>
#include <cfloat>

// ---------------------------------------------------------------------------
// VQ-VAE forward on gfx1250 (MI455X), fp32 WMMA (V_WMMA_F32_16X16X4_F32).
// Shapes: B=32768, IN=256, HID=1024, LATENT=128, K=1024.
// Software-pipelined: next K-chunk's global loads are issued while the
// current chunk's WMMAs run from LDS; +global_prefetch one chunk further out.
// ---------------------------------------------------------------------------

#define BB      32768
#define IN_DIM  256
#define HID     1024
#define LATENT  128
#define NCODE   1024

typedef float v2f __attribute__((ext_vector_type(2)));
typedef float v8f __attribute__((ext_vector_type(8)));

// LDS strides (floats) chosen for 8/16B alignment + bank spread.
#define AS_STRIDE 18     // A tile: [128 rows][16 k], row stride 18 (8B aligned)
#define WS_STRIDE 132    // B tile: [16 k][128 n], row stride 132 (16B aligned, conflict-free)

// ---------------------------------------------------------------------------
// GEMM: Y[M,N] = act(X[M,K] @ W[K,N] + bias[N]),  ACT: 0=identity, 1=tanh
// Block: 256 thr (8 waves). Block tile 128x128, wave tile 32x64 (2x4 WMMA tiles).
// K staged through LDS in chunks of 16; inner WMMA K-step = 4 (fp32).
// ---------------------------------------------------------------------------
template <int ACT>
__global__ __launch_bounds__(256) void gemm_bias_act(
    const float* __restrict__ X, const float* __restrict__ W,
    const float* __restrict__ bias, float* __restrict__ Y,
    int M, int Kd, int Nd)
{
    __shared__ float As[128 * AS_STRIDE];
    __shared__ float Ws[16 * WS_STRIDE];

    const int tid  = threadIdx.x;
    const int w    = tid >> 5;
    const int lane = tid & 31;
    const int l15  = lane & 15;
    const int hi   = lane >> 4;          // 0: lanes 0-15, 1: lanes 16-31

    const int blockM = blockIdx.y * 128;
    const int blockN = blockIdx.x * 128;
    const int waveM  = (w >> 1) * 32;    // 0,32,64,96
    const int waveN  = (w & 1) * 64;     // 0,64

    // Per-thread staging coordinates (2 float4 for A tile, 2 for W tile).
    const int aRow0 = (tid      ) >> 2, aC0 = ((tid      ) & 3) * 4;
    const int aRow1 = (tid + 256) >> 2, aC1 = ((tid + 256) & 3) * 4;
    const int wRow0 = (tid      ) >> 5, wC0 = ((tid      ) & 31) * 4;
    const int wRow1 = (tid + 256) >> 5, wC1 = ((tid + 256) & 31) * 4;

    v8f acc[2][4];
#pragma unroll
    for (int ms = 0; ms < 2; ++ms)
#pragma unroll
        for (int ns = 0; ns < 4; ++ns)
            acc[ms][ns] = {};

    // Prologue: fetch chunk 0 into registers.
    float4 pa0 = *(const float4*)(X + (size_t)(blockM + aRow0) * Kd + aC0);
    float4 pa1 = *(const float4*)(X + (size_t)(blockM + aRow1) * Kd + aC1);
    float4 pw0 = *(const float4*)(W + (size_t)wRow0 * Nd + blockN + wC0);
    float4 pw1 = *(const float4*)(W + (size_t)wRow1 * Nd + blockN + wC1);

    for (int kc = 0; kc < Kd; kc += 16) {
        __syncthreads();            // previous chunk fully consumed
        // Commit staged registers to LDS.
        {
            float* d0 = &As[aRow0 * AS_STRIDE + aC0];
            ((float2*)d0)[0] = make_float2(pa0.x, pa0.y);
            ((float2*)d0)[1] = make_float2(pa0.z, pa0.w);
            float* d1 = &As[aRow1 * AS_STRIDE + aC1];
            ((float2*)d1)[0] = make_float2(pa1.x, pa1.y);
            ((float2*)d1)[1] = make_float2(pa1.z, pa1.w);
            *(float4*)&Ws[wRow0 * WS_STRIDE + wC0] = pw0;
            *(float4*)&Ws[wRow1 * WS_STRIDE + wC1] = pw1;
        }
        __syncthreads();

        // Issue next chunk's global loads now: they overlap the WMMA phase.
        const int kn = kc + 16;
        if (kn < Kd) {
            pa0 = *(const float4*)(X + (size_t)(blockM + aRow0) * Kd + kn + aC0);
            pa1 = *(const float4*)(X + (size_t)(blockM + aRow1) * Kd + kn + aC1);
            pw0 = *(const float4*)(W + (size_t)(kn + wRow0) * Nd + blockN + wC0);
            pw1 = *(const float4*)(W + (size_t)(kn + wRow1) * Nd + blockN + wC1);
            if (kn + 16 < Kd) {     // L2 prefetch one more chunk out
                __builtin_prefetch(X + (size_t)(blockM + aRow0) * Kd + kn + 16 + aC0, 0, 3);
                __builtin_prefetch(W + (size_t)(kn + 16 + wRow0) * Nd + blockN + wC0, 0, 3);
            }
        }

        // 4 WMMA K-steps of 4 over the 16-deep LDS tile.
#pragma unroll
        for (int kk = 0; kk < 16; kk += 4) {
            v2f a[2], b[4];
            // A 16x4 frag: v0 = K=kk(+2 for hi lanes), v1 = K=kk+1(+2).
#pragma unroll
            for (int ms = 0; ms < 2; ++ms)
                a[ms] = *(const v2f*)&As[(waveM + ms * 16 + l15) * AS_STRIDE + kk + hi * 2];
            // B 4x16 frag: v0 = row K=kk(+2 hi), v1 = row K=kk+1(+2 hi), col = n0+l15.
#pragma unroll
            for (int ns = 0; ns < 4; ++ns) {
                const float* bp = &Ws[(kk + hi * 2) * WS_STRIDE + waveN + ns * 16 + l15];
                b[ns] = (v2f){ bp[0], bp[WS_STRIDE] };
            }
#pragma unroll
            for (int ms = 0; ms < 2; ++ms)
#pragma unroll
                for (int ns = 0; ns < 4; ++ns)
                    acc[ms][ns] = __builtin_amdgcn_wmma_f32_16x16x4_f32(
                        false, a[ms], false, b[ns], (short)0, acc[ms][ns], false, false);
        }
    }

    // Epilogue: bias + activation + store (C/D layout: VGPR r -> rows r / r+8).
#pragma unroll
    for (int ns = 0; ns < 4; ++ns) {
        const int col = blockN + waveN + ns * 16 + l15;
        const float bv = bias[col];
#pragma unroll
        for (int ms = 0; ms < 2; ++ms) {
#pragma unroll
            for (int r = 0; r < 8; ++r) {
                const int row = blockM + waveM + ms * 16 + r + hi * 8;
                float y = acc[ms][ns][r] + bv;
                if (ACT) y = tanhf(y);
                Y[(size_t)row * Nd + col] = y;
            }
        }
    }
}

// ---------------------------------------------------------------------------
// e2[k] = sum_j emb[k][j]^2
// ---------------------------------------------------------------------------
__global__ __launch_bounds__(256) void codebook_norms(
    const float* __restrict__ emb, float* __restrict__ e2)
{
    const int k = blockIdx.x * blockDim.x + threadIdx.x;
    if (k < NCODE) {
        float s = 0.f;
#pragma unroll 4
        for (int j = 0; j < LATENT; ++j) {
            const float v = emb[(size_t)k * LATENT + j];
            s += v * v;
        }
        e2[k] = s;
    }
}

// ---------------------------------------------------------------------------
// VQ: per-row argmin_k (e2[k] - 2 z.e_k) via WMMA scores, then z_q = emb[idx]
// and loss accumulation sum((z - z_q)^2).
// Block: 256 thr (8 waves), 128 rows/block (16 rows/wave). Codebook tiles of
// 16 codes staged in LDS (shared by all waves), pipelined against the WMMAs.
// ---------------------------------------------------------------------------
__global__ __launch_bounds__(256) void vq_kernel(
    const float* __restrict__ Z, const float* __restrict__ emb,
    const float* __restrict__ e2, float* __restrict__ zq_out,
    float* __restrict__ loss_accum)
{
    __shared__ float Es[16 * WS_STRIDE];   // embT tile: [16 codes][128 k]
    __shared__ int   sIdx[128];

    const int tid  = threadIdx.x;
    const int w    = tid >> 5;
    const int lane = tid & 31;
    const int l15  = lane & 15;
    const int hi   = lane >> 4;
    const int m0   = blockIdx.x * 128 + w * 16;

    const int eRow0 = (tid      ) >> 5, eC0 = ((tid      ) & 31) * 4;
    const int eRow1 = (tid + 256) >> 5, eC1 = ((tid + 256) & 31) * 4;

    // Preload this wave's 16 z-rows as 32 fp32 A-fragments (K=128 total).
    v2f a[32];
    {
        const float* zr = Z + (size_t)(m0 + l15) * LATENT + hi * 2;
#pragma unroll
        for (int t = 0; t < 32; ++t)
            a[t] = *(const v2f*)(zr + 4 * t);
    }

    float minv[8];
    int   mini[8];
#pragma unroll
    for (int r = 0; r < 8; ++r) { minv[r] = FLT_MAX; mini[r] = 0; }

    // Prologue: fetch codebook tile 0 into registers.
    float4 pe0 = *(const float4*)(emb + (size_t)eRow0 * LATENT + eC0);
    float4 pe1 = *(const float4*)(emb + (size_t)eRow1 * LATENT + eC1);

    for (int n0 = 0; n0 < NCODE; n0 += 16) {
        __syncthreads();
        *(float4*)&Es[eRow0 * WS_STRIDE + eC0] = pe0;
        *(float4*)&Es[eRow1 * WS_STRIDE + eC1] = pe1;
        __syncthreads();

        // Next tile's global loads overlap this tile's WMMAs.
        if (n0 + 16 < NCODE) {
            pe0 = *(const float4*)(emb + (size_t)(n0 + 16 + eRow0) * LATENT + eC0);
            pe1 = *(const float4*)(emb + (size_t)(n0 + 16 + eRow1) * LATENT + eC1);
        }

        const float e2v = e2[n0 + l15];

        v8f s = {};
#pragma unroll
        for (int t = 0; t < 32; ++t) {
            // B frag: column n0+l15's latent values k=4t.. (B=embT is 128x16).
            const v2f b = *(const v2f*)&Es[l15 * WS_STRIDE + 4 * t + hi * 2];
            s = __builtin_amdgcn_wmma_f32_16x16x4_f32(
                false, a[t], false, b, (short)0, s, false, false);
        }
#pragma unroll
        for (int r = 0; r < 8; ++r) {
            const float d = e2v - 2.f * s[r];
            if (d < minv[r]) { minv[r] = d; mini[r] = n0 + l15; }
        }
    }

    // Min-reduce across the 16 lanes sharing each row (xor masks stay in-half).
#pragma unroll
    for (int r = 0; r < 8; ++r) {
#pragma unroll
        for (int off = 1; off < 16; off <<= 1) {
            const float ov = __shfl_xor(minv[r], off, 32);
            const int   oi = __shfl_xor(mini[r], off, 32);
            if (ov < minv[r] || (ov == minv[r] && oi < mini[r])) {
                minv[r] = ov; mini[r] = oi;
            }
        }
        if (l15 == 0) sIdx[w * 16 + r + hi * 8] = mini[r];
    }
    __syncthreads();

    // Gather z_q = emb[idx] and accumulate the MSE numerator.
    const int row  = tid >> 1;                 // 0..127
    const int half = tid & 1;
    const int gRow = blockIdx.x * 128 + row;
    const int code = sIdx[row];
    float lsum = 0.f;
    const int j0 = half * 64;
#pragma unroll 4
    for (int j = j0; j < j0 + 64; ++j) {
        const float q  = emb[(size_t)code * LATENT + j];
        const float zv = Z[(size_t)gRow * LATENT + j];
        zq_out[(size_t)gRow * LATENT + j] = q;
        const float d = zv - q;
        lsum += d * d;
    }
    atomicAdd(loss_accum, lsum);
}

__global__ void finalize_loss(const float* __restrict__ accum, float* __restrict__ loss_out)
{
    if (threadIdx.x == 0 && blockIdx.x == 0)
        *loss_out = 1.25f * (*accum) / (float)((size_t)BB * LATENT);
}

// ---------------------------------------------------------------------------
// Host launcher
// ---------------------------------------------------------------------------
extern "C" void kernel_launch(void* const* d_in, const int* in_sizes, int n_in,
                              void* d_out, int out_size, void* d_ws, size_t ws_size,
                              hipStream_t stream)
{
    // setup_inputs order: state, action, eW0, eb0, eW1, eb1, eW2, eb2, emb,
    //                     dW0, db0, dW1, db1, dW2, db2
    const float* action = (const float*)d_in[1];
    const float* eW0 = (const float*)d_in[2];  const float* eb0 = (const float*)d_in[3];
    const float* eW1 = (const float*)d_in[4];  const float* eb1 = (const float*)d_in[5];
    const float* eW2 = (const float*)d_in[6];  const float* eb2 = (const float*)d_in[7];
    const float* emb = (const float*)d_in[8];
    const float* dW0 = (const float*)d_in[9];  const float* db0 = (const float*)d_in[10];
    const float* dW1 = (const float*)d_in[11]; const float* db1 = (const float*)d_in[12];
    const float* dW2 = (const float*)d_in[13]; const float* db2 = (const float*)d_in[14];

    // Output layout: recon [B,256] | z [B,128] | z_q [B,128] | vq_loss [1]
    float* recon = (float*)d_out;
    float* z     = recon + (size_t)BB * IN_DIM;
    float* zq    = z     + (size_t)BB * LATENT;
    float* loss  = zq    + (size_t)BB * LATENT;

    // Workspace: two [B,HID] activation buffers (ping-pong), e2[K], accumulator.
    float* Ha  = (float*)d_ws;
    float* Hb  = Ha + (size_t)BB * HID;
    float* e2  = Hb + (size_t)BB * HID;
    float* acc = e2 + NCODE;

    hipMemsetAsync(acc, 0, sizeof(float), stream);
    codebook_norms<<<NCODE / 256, 256, 0, stream>>>(emb, e2);

    const dim3 blk(256);
    // Encoder
    gemm_bias_act<1><<<dim3(HID / 128,    BB / 128), blk, 0, stream>>>(action, eW0, eb0, Ha, BB, IN_DIM, HID);
    gemm_bias_act<1><<<dim3(HID / 128,    BB / 128), blk, 0, stream>>>(Ha,     eW1, eb1, Hb, BB, HID,    HID);
    gemm_bias_act<0><<<dim3(LATENT / 128, BB / 128), blk, 0, stream>>>(Hb,     eW2, eb2, z,  BB, HID,    LATENT);
    // Vector quantization (writes z_q, loss numerator)
    vq_kernel<<<BB / 128, blk, 0, stream>>>(z, emb, e2, zq, acc);
    // Decoder (reads z_q)
    gemm_bias_act<1><<<dim3(HID / 128,    BB / 128), blk, 0, stream>>>(zq, dW0, db0, Ha,    BB, LATENT, HID);
    gemm_bias_act<1><<<dim3(HID / 128,    BB / 128), blk, 0, stream>>>(Ha, dW1, db1, Hb,    BB, HID,    HID);
    gemm_bias_act<0><<<dim3(IN_DIM / 128, BB / 128), blk, 0, stream>>>(Hb, dW2, db2, recon, BB, HID,    IN_DIM);

    finalize_loss<<<1, 1, 0, stream>>>(acc, loss);
}